// SSIM_1726576857610
// MI455X (gfx1250) — compile-verified
//
#include <hip/hip_runtime.h>
#include <cstdint>

typedef __attribute__((ext_vector_type(2))) float v2f;
typedef __attribute__((ext_vector_type(8))) float v8f;

#define IMG_H 384
#define IMG_W 512
#define NPLANES 48            // 16 batch * 3 channels
#define TILES_Y 28            // ceil(384/14)
#define TILES_X 37            // ceil(512/14)
#define WAVES_PER_BLOCK 8
#define LDS_STRIDE 17         // padded row stride (floats) to avoid bank conflicts

__device__ __forceinline__ int refl1(int i, int n) {
    // reflect-pad width 1 (mirror without edge repeat); only single step needed
    if (i < 0) i = -i;
    if (i >= n) i = 2 * n - 2 - i;
    return i;
}

__device__ __forceinline__ v8f wmma4(v2f a, v2f b, v8f c) {
    // D = A(16x4) * B(4x16) + C, fp32, wave32
    return __builtin_amdgcn_wmma_f32_16x16x4_f32(
        /*neg_a=*/false, a, /*neg_b=*/false, b,
        /*c_mod=*/(short)0, c, /*reuse_a=*/false, /*reuse_b=*/false);
}

// Vertical 3-window sum: S = A * T, where T arrives in WMMA C/D layout.
// Convert T (D layout) -> B fragments via half-wave swaps + selects, then 4 WMMAs.
__device__ __forceinline__ v8f pool_vert(v8f T, const v2f wband[4], bool hi) {
    float Td[8], Sw[8];
#pragma unroll
    for (int r = 0; r < 8; ++r) {
        Td[r] = T[r];
        Sw[r] = __shfl_xor(T[r], 16, 32);   // lanes L <-> L^16
    }
    v2f B0, B1, B2, B3;
    B0[0] = hi ? Sw[2] : Td[0];  B0[1] = hi ? Sw[3] : Td[1];   // K = 0,1 | 2,3
    B1[0] = hi ? Sw[6] : Td[4];  B1[1] = hi ? Sw[7] : Td[5];   // K = 4,5 | 6,7
    B2[0] = hi ? Td[2] : Sw[0];  B2[1] = hi ? Td[3] : Sw[1];   // K = 8,9 | 10,11
    B3[0] = hi ? Td[6] : Sw[4];  B3[1] = hi ? Td[7] : Sw[5];   // K = 12,13 | 14,15
    v8f S = {};
    S = wmma4(wband[0], B0, S);
    S = wmma4(wband[1], B1, S);
    S = wmma4(wband[2], B2, S);
    S = wmma4(wband[3], B3, S);
    return S;
}

__global__ __launch_bounds__(256) void ssim_tile_kernel(
    const float* __restrict__ x, const float* __restrict__ y,
    float* __restrict__ out)
{
    __shared__ float lds[WAVES_PER_BLOCK][2][16 * LDS_STRIDE];

    const int lane = threadIdx.x & 31;
    const int wv   = threadIdx.x >> 5;
    const bool hi  = lane >= 16;
    const int m    = lane & 15;

    const int tile  = blockIdx.x * WAVES_PER_BLOCK + wv;
    const int plane = tile / (TILES_Y * TILES_X);
    const int rem   = tile % (TILES_Y * TILES_X);
    int oy = (rem / TILES_X) * 14; if (oy > IMG_H - 14) oy = IMG_H - 14;
    int ox = (rem % TILES_X) * 14; if (ox > IMG_W - 14) ox = IMG_W - 14;

    const size_t planeOff = (size_t)plane * IMG_H * IMG_W;
    const float* xp = x + planeOff;
    const float* yp = y + planeOff;

    // ---- async stage 16x16 tiles (with reflect halo) into per-wave LDS ----
    const uint32_t ldsX = (uint32_t)(uintptr_t)&lds[wv][0][0];  // low 32 bits of
    const uint32_t ldsY = (uint32_t)(uintptr_t)&lds[wv][1][0];  // flat = LDS offset
#pragma unroll
    for (int it = 0; it < 8; ++it) {
        const int e  = it * 32 + lane;          // element 0..255
        const int r  = e >> 4;
        const int c  = e & 15;
        const int iy = refl1(oy + r - 1, IMG_H);
        const int ix = refl1(ox + c - 1, IMG_W);
        const uint32_t loff = (uint32_t)((r * LDS_STRIDE + c) * 4);
        const uint64_t gax = (uint64_t)(uintptr_t)(xp + (size_t)iy * IMG_W + ix);
        const uint64_t gay = (uint64_t)(uintptr_t)(yp + (size_t)iy * IMG_W + ix);
        const uint32_t dX = ldsX + loff;
        const uint32_t dY = ldsY + loff;
        asm volatile("global_load_async_to_lds_b32 %0, %1, off"
                     :: "v"(dX), "v"(gax) : "memory");
        asm volatile("global_load_async_to_lds_b32 %0, %1, off"
                     :: "v"(dY), "v"(gay) : "memory");
    }
    asm volatile("s_wait_asynccnt 0" ::: "memory");

    // ---- read Q in WMMA A-fragment layout straight from LDS ----
    const float* lx = &lds[wv][0][0];
    const float* ly = &lds[wv][1][0];
    v2f xa[4], ya[4];
#pragma unroll
    for (int c = 0; c < 4; ++c) {
        const int k0   = 4 * c + (hi ? 2 : 0);
        const int base = m * LDS_STRIDE + k0;
        xa[c][0] = lx[base];     xa[c][1] = lx[base + 1];
        ya[c][0] = ly[base];     ya[c][1] = ly[base + 1];
    }

    // Band matrix A(i,k)=1 for k in [i, i+2], rows i<=13.
    // Same per-lane values serve as A-fragment (stage 2) and B-fragment of A^T (stage 1).
    v2f wband[4];
#pragma unroll
    for (int c = 0; c < 4; ++c) {
        const int k0 = 4 * c + (hi ? 2 : 0);
        wband[c][0] = (m <= 13 && k0     >= m && k0     <= m + 2) ? 1.0f : 0.0f;
        wband[c][1] = (m <= 13 && k0 + 1 >= m && k0 + 1 <= m + 2) ? 1.0f : 0.0f;
    }

    // ---- stage 1: horizontal 3-window sums  T = Q * A^T  (5 quantities) ----
    v8f Tx = {}, Ty = {}, Txx = {}, Tyy = {}, Txy = {};
#pragma unroll
    for (int c = 0; c < 4; ++c) {
        const v2f qx = xa[c], qy = ya[c];
        const v2f qxx = qx * qx, qyy = qy * qy, qxy = qx * qy;
        Tx  = wmma4(qx,  wband[c], Tx);
        Ty  = wmma4(qy,  wband[c], Ty);
        Txx = wmma4(qxx, wband[c], Txx);
        Tyy = wmma4(qyy, wband[c], Tyy);
        Txy = wmma4(qxy, wband[c], Txy);
    }

    // ---- stage 2: vertical 3-window sums  S = A * T ----
    const v8f Sx  = pool_vert(Tx,  wband, hi);
    const v8f Sy  = pool_vert(Ty,  wband, hi);
    const v8f Sxx = pool_vert(Txx, wband, hi);
    const v8f Syy = pool_vert(Tyy, wband, hi);
    const v8f Sxy = pool_vert(Txy, wband, hi);

    // ---- pointwise SSIM + clip, store valid 14x14 region ----
    const float C1 = 0.01f * 0.01f;
    const float C2 = 0.03f * 0.03f;
    const float inv9 = 1.0f / 9.0f;
    float* op = out + planeOff;
#pragma unroll
    for (int r = 0; r < 8; ++r) {
        const float mux = Sx[r] * inv9;
        const float muy = Sy[r] * inv9;
        const float sxx = Sxx[r] * inv9 - mux * mux;
        const float syy = Syy[r] * inv9 - muy * muy;
        const float sxy = Sxy[r] * inv9 - mux * muy;
        const float num = (2.0f * mux * muy + C1) * (2.0f * sxy + C2);
        const float den = (mux * mux + muy * muy + C1) * (sxx + syy + C2);
        float v = (1.0f - num / den) * 0.5f;
        v = fminf(fmaxf(v, 0.0f), 1.0f);
        const int i = r + (hi ? 8 : 0);
        if (i < 14 && m < 14)
            op[(size_t)(oy + i) * IMG_W + (ox + m)] = v;
    }
}

extern "C" void kernel_launch(void* const* d_in, const int* in_sizes, int n_in,
                              void* d_out, int out_size, void* d_ws, size_t ws_size,
                              hipStream_t stream) {
    (void)in_sizes; (void)n_in; (void)out_size; (void)d_ws; (void)ws_size;
    const float* x = (const float*)d_in[0];
    const float* y = (const float*)d_in[1];
    float* out = (float*)d_out;
    const int total_tiles = NPLANES * TILES_Y * TILES_X;      // 49728
    const int blocks = total_tiles / WAVES_PER_BLOCK;         // 6216, exact
    hipLaunchKernelGGL(ssim_tile_kernel, dim3(blocks), dim3(256), 0, stream,
                       x, y, out);
}